// CFContrastiveLoss_38285338477148
// MI455X (gfx1250) — compile-verified
//
#include <hip/hip_runtime.h>
#include <hip/hip_bf16.h>
#include <math.h>

typedef __attribute__((ext_vector_type(2))) float v2f;
typedef __attribute__((ext_vector_type(8))) float v8f;

#define N_OPT   16
#define DIM     768
#define MARGIN  0.3f
#define NEG_INF -1e9f
#define WAVES_PER_BLOCK 8

// One wave32 per sample. sim = E * E^T via V_WMMA_F32_16X16X4_F32,
// exploiting that A(16x4) and B(4x16) lane layouts coincide for E and E^T:
// lane L holds E[L%16][k + 2*(L>=16) + {0,1}]  -> same v2f feeds SRC0 and SRC1.
__global__ void __launch_bounds__(256)
cf_contrastive_loss_kernel(const float* __restrict__ emb,
                           const int* __restrict__ labels,
                           float* __restrict__ ws)
{
    __shared__ float s_norm[WAVES_PER_BLOCK][N_OPT]; // raw squared norms (Gram diag)
    __shared__ int   s_lab [WAVES_PER_BLOCK][N_OPT];

    const int lane  = threadIdx.x & 31;
    const int warp  = threadIdx.x >> 5;
    const int b     = blockIdx.x * WAVES_PER_BLOCK + warp;   // sample index

    const int row   = lane & 15;           // matrix row (A) == matrix col (B)
    const int khalf = (lane >> 4) << 1;    // 0 for lanes 0-15, 2 for lanes 16-31
    const float* base = emb + (size_t)b * (N_OPT * DIM) + (size_t)row * DIM + khalf;

    // ---- Gram matrix accumulation: 192 x v_wmma_f32_16x16x4_f32 ----
    v8f acc = {};
#pragma unroll 4
    for (int k = 0; k < DIM; k += 4) {
        v2f a = *(const v2f*)(base + k);   // 8B load, K-pair for this lane-half
        acc = __builtin_amdgcn_wmma_f32_16x16x4_f32(
                  /*neg_a=*/false, a, /*neg_b=*/false, a,
                  /*c_mod=*/(short)0, acc, /*reuse_a=*/false, /*reuse_b=*/false);
    }

    // ---- stage labels + Gram diagonal (squared norms) through LDS ----
    if (lane < N_OPT) s_lab[warp][lane] = labels[b * N_OPT + lane];

    const int mbase = (lane >> 4) << 3;    // rows covered by this lane-half
#pragma unroll
    for (int v = 0; v < 8; ++v) {
        int m = mbase + v;                 // C/D layout: row = v + 8*(lane>=16)
        if ((lane & 15) == m)              // column index == row -> diagonal owner
            s_norm[warp][m] = acc[v];
    }
    __syncthreads();

    // ---- validity: sample counts only if it has >=1 pos and >=1 neg ----
    int hp = 0, hn = 0;
#pragma unroll
    for (int i = 0; i < N_OPT; ++i) {
        int l = s_lab[warp][i];
        hp |= (l == 1);
        hn |= (l == 0);
    }
    const bool valid = (hp != 0) && (hn != 0);

    const int   lab_col = s_lab[warp][lane & 15];
    const float dc      = s_norm[warp][lane & 15];
    const float nc      = fmaxf(sqrtf(fmaxf(dc, 0.0f)), 1e-12f);

    float local_sum = 0.0f;
    float local_cnt = 0.0f;

#pragma unroll
    for (int v = 0; v < 8; ++v) {
        const int   m  = mbase + v;
        const float dr = s_norm[warp][m];
        const float nr = fmaxf(sqrtf(fmaxf(dr, 0.0f)), 1e-12f);
        const float cs = acc[v] / (nr * nc);          // cosine similarity
        float mx = (lab_col == 0) ? cs : NEG_INF;     // mask to negative columns
        // max over the 16 columns of this row (16-lane butterfly)
#pragma unroll
        for (int off = 1; off < 16; off <<= 1)
            mx = fmaxf(mx, __shfl_xor(mx, off, 16));
        if ((lane & 15) == 0) {                       // one lane per row-group
            if (valid && s_lab[warp][m] == 1) {
                local_sum += fmaxf(mx + MARGIN, 0.0f);
                local_cnt += 1.0f;
            }
        }
    }

    if ((lane & 15) == 0) {                           // lanes 0 and 16
        atomicAdd(&ws[0], local_sum);
        atomicAdd(&ws[1], local_cnt);
    }
}

__global__ void cf_init_ws(float* __restrict__ ws) {
    if (threadIdx.x < 2) ws[threadIdx.x] = 0.0f;
}

__global__ void cf_finalize(const float* __restrict__ ws, float* __restrict__ out) {
    if (threadIdx.x == 0) out[0] = ws[0] / fmaxf(ws[1], 1.0f);
}

extern "C" void kernel_launch(void* const* d_in, const int* in_sizes, int n_in,
                              void* d_out, int out_size, void* d_ws, size_t ws_size,
                              hipStream_t stream) {
    (void)in_sizes; (void)n_in; (void)out_size; (void)ws_size;
    const float* emb    = (const float*)d_in[0];   // [8192,16,768] fp32
    const int*   labels = (const int*)d_in[1];     // [8192,16] int
    float*       ws     = (float*)d_ws;            // [0]=loss_sum, [1]=count
    float*       out    = (float*)d_out;           // scalar

    cf_init_ws<<<1, 32, 0, stream>>>(ws);
    const int samples = 8192;
    cf_contrastive_loss_kernel<<<samples / WAVES_PER_BLOCK, 32 * WAVES_PER_BLOCK, 0, stream>>>(
        emb, labels, ws);
    cf_finalize<<<1, 32, 0, stream>>>(ws, out);
}